// Downsampling_28278064677298
// MI455X (gfx1250) — compile-verified
//
#include <hip/hip_runtime.h>
#include <hip/hip_bf16.h>
#include <math.h>

#define NPTS   16384
#define MPTS   4096       // NPTS / STRIDE
#define DIN    64
#define DOUT   128
#define KNN    16
#define KPAD   68         // 3 + 64 = 67, padded to multiple of 4
#define BN_EPS 1e-5f

typedef __attribute__((ext_vector_type(2))) float v2f;
typedef __attribute__((ext_vector_type(8))) float v8f;

// ---------------------------------------------------------------------------
// Kernel 1: Furthest point sampling. Single workgroup of 1024 threads.
// Per-thread min-dist cache in registers (16 points/thread). Two-stage argmax:
// wave32 shuffle reduction, then LDS reduction across the 32 waves.
// ---------------------------------------------------------------------------
#define FPS_T 1024
#define PER_T (NPTS / FPS_T)   // 16

__global__ __launch_bounds__(FPS_T)
void fps_kernel(const float* __restrict__ p, int* __restrict__ sidx) {
    __shared__ float s_wval[FPS_T / 32];
    __shared__ int   s_widx[FPS_T / 32];
    __shared__ float s_last[3];

    const int t = threadIdx.x;
    const int lane = t & 31;
    const int wid  = t >> 5;

    float dists[PER_T];
#pragma unroll
    for (int j = 0; j < PER_T; ++j) dists[j] = 3.4e38f;

    if (t == 0) {
        sidx[0]   = 0;
        s_last[0] = p[0];
        s_last[1] = p[1];
        s_last[2] = p[2];
    }
    __syncthreads();

    for (int i = 1; i < MPTS; ++i) {
        const float lx = s_last[0], ly = s_last[1], lz = s_last[2];
        float best = -1.0f;
        int   bidx = 0;
#pragma unroll
        for (int j = 0; j < PER_T; ++j) {
            const int idx = j * FPS_T + t;
            const float dx = p[idx * 3 + 0] - lx;
            const float dy = p[idx * 3 + 1] - ly;
            const float dz = p[idx * 3 + 2] - lz;
            const float d  = dx * dx + dy * dy + dz * dz;
            const float dm = fminf(dists[j], d);
            dists[j] = dm;
            if (dm > best) { best = dm; bidx = idx; }   // ascending idx -> first-max kept
        }
        // wave32 argmax reduction (tie -> smaller index, matching argmax semantics)
#pragma unroll
        for (int off = 16; off > 0; off >>= 1) {
            const float ov = __shfl_down(best, off, 32);
            const int   oi = __shfl_down(bidx, off, 32);
            if (ov > best || (ov == best && oi < bidx)) { best = ov; bidx = oi; }
        }
        if (lane == 0) { s_wval[wid] = best; s_widx[wid] = bidx; }
        __syncthreads();
        if (t < 32) {
            best = s_wval[t];
            bidx = s_widx[t];
#pragma unroll
            for (int off = 16; off > 0; off >>= 1) {
                const float ov = __shfl_down(best, off, 32);
                const int   oi = __shfl_down(bidx, off, 32);
                if (ov > best || (ov == best && oi < bidx)) { best = ov; bidx = oi; }
            }
            if (t == 0) {
                sidx[i]   = bidx;
                s_last[0] = p[bidx * 3 + 0];
                s_last[1] = p[bidx * 3 + 1];
                s_last[2] = p[bidx * 3 + 2];
            }
        }
        __syncthreads();
    }
}

// ---------------------------------------------------------------------------
// Kernel 2: per sampled point: kNN over all N (register top-16 insertion),
// relative-coord normalization, [pj|xj] max-pool -> feat[M][KPAD] (k=67 zero).
// Also emits n_p into d_out. One thread per sampled point; waves stream p[j]
// in lockstep (broadcast loads, p fully L2-resident).
// ---------------------------------------------------------------------------
__global__ __launch_bounds__(256)
void knn_group_kernel(const float* __restrict__ p, const float* __restrict__ x,
                      const int* __restrict__ sidx,
                      float* __restrict__ feat, float* __restrict__ np_out) {
    const int m = blockIdx.x * blockDim.x + threadIdx.x;
    if (m >= MPTS) return;

    const int   si = sidx[m];
    const float cx = p[si * 3 + 0];
    const float cy = p[si * 3 + 1];
    const float cz = p[si * 3 + 2];
    np_out[m * 3 + 0] = cx;
    np_out[m * 3 + 1] = cy;
    np_out[m * 3 + 2] = cz;

    float nd[KNN];
    int   ni[KNN];
#pragma unroll
    for (int k = 0; k < KNN; ++k) { nd[k] = 3.4e38f; ni[k] = 0; }

    for (int j = 0; j < NPTS; ++j) {
        const float dx = p[j * 3 + 0] - cx;
        const float dy = p[j * 3 + 1] - cy;
        const float dz = p[j * 3 + 2] - cz;
        const float d2 = dx * dx + dy * dy + dz * dz;
        if (d2 < nd[KNN - 1]) {
            int pos = KNN - 1;
            while (pos > 0 && nd[pos - 1] > d2) {
                nd[pos] = nd[pos - 1];
                ni[pos] = ni[pos - 1];
                --pos;
            }
            nd[pos] = d2;
            ni[pos] = j;
        }
    }

    // relative coords + max neighbor norm
    float pjx[KNN], pjy[KNN], pjz[KNN];
    float maxn2 = 0.0f;
#pragma unroll
    for (int k = 0; k < KNN; ++k) {
        const int j = ni[k];
        const float dx = p[j * 3 + 0] - cx;
        const float dy = p[j * 3 + 1] - cy;
        const float dz = p[j * 3 + 2] - cz;
        pjx[k] = dx; pjy[k] = dy; pjz[k] = dz;
        maxn2 = fmaxf(maxn2, dx * dx + dy * dy + dz * dz);
    }
    const float inv = 1.0f / (sqrtf(maxn2) + 1e-8f);

    float mx = -3.4e38f, my = -3.4e38f, mz = -3.4e38f;
#pragma unroll
    for (int k = 0; k < KNN; ++k) {
        mx = fmaxf(mx, pjx[k] * inv);
        my = fmaxf(my, pjy[k] * inv);
        mz = fmaxf(mz, pjz[k] * inv);
    }
    float* fr = feat + (size_t)m * KPAD;
    fr[0] = mx; fr[1] = my; fr[2] = mz;

    for (int f = 0; f < DIN; ++f) {
        float mv = -3.4e38f;
#pragma unroll
        for (int k = 0; k < KNN; ++k)
            mv = fmaxf(mv, x[(size_t)ni[k] * DIN + f]);
        fr[3 + f] = mv;
    }
    fr[KPAD - 1] = 0.0f;   // zero-pad K dim (67 -> 68)
}

// ---------------------------------------------------------------------------
// Kernel 3: pad W [67][128] -> Wp [68][128] (row 67 zero).
// ---------------------------------------------------------------------------
__global__ __launch_bounds__(256)
void pad_w_kernel(const float* __restrict__ W, float* __restrict__ Wp) {
    const int i = blockIdx.x * blockDim.x + threadIdx.x;
    if (i >= KPAD * DOUT) return;
    const int r = i / DOUT;
    Wp[i] = (r < KPAD - 1) ? W[i] : 0.0f;
}

// ---------------------------------------------------------------------------
// Kernel 4: h = feat[4096x68] @ Wp[68x128] using V_WMMA_F32_16X16X4_F32.
// One block (8 waves) per 16-row M tile; wave w owns N-tile w (16 cols).
// A frag (16x4 f32, 2 VGPR): v0 = K+{0|2}, v1 = K+{1|3} per lane half.
// B frag (4x16 f32, 2 VGPR): row-striped across lanes, K and K+2 halves.
// C/D: 8 VGPRs, row = v + 8*(lane>=16), col = lane&15.
// ---------------------------------------------------------------------------
__global__ __launch_bounds__(256)
void gemm_wmma_kernel(const float* __restrict__ feat, const float* __restrict__ Wp,
                      float* __restrict__ h) {
    const int wave = threadIdx.x >> 5;          // 0..7 -> N tile
    const int lane = threadIdx.x & 31;
    const int mt   = blockIdx.x;                // 0..255
    const int row  = lane & 15;
    const int hi   = lane >> 4;                 // 0 or 1

    const float* Arow = feat + (size_t)(mt * 16 + row) * KPAD;
    const int    ncol = wave * 16 + row;        // B/N index for this lane

    v8f c = {};
#pragma unroll
    for (int k0 = 0; k0 < KPAD; k0 += 4) {
        v2f a, b;
        a.x = Arow[k0 + 2 * hi + 0];
        a.y = Arow[k0 + 2 * hi + 1];
        b.x = Wp[(size_t)(k0 + 2 * hi + 0) * DOUT + ncol];
        b.y = Wp[(size_t)(k0 + 2 * hi + 1) * DOUT + ncol];
        c = __builtin_amdgcn_wmma_f32_16x16x4_f32(
                /*neg_a=*/false, a, /*neg_b=*/false, b,
                /*c_mod=*/(short)0, c, /*reuse_a=*/false, /*reuse_b=*/false);
    }

    float* hp = h + (size_t)(mt * 16) * DOUT + wave * 16;
#pragma unroll
    for (int v = 0; v < 8; ++v)
        hp[(size_t)(v + 8 * hi) * DOUT + row] = c[v];
}

// ---------------------------------------------------------------------------
// Kernel 5: batch-norm statistics per channel (bias b cancels in batch BN).
// ---------------------------------------------------------------------------
__global__ __launch_bounds__(DOUT)
void bn_stats_kernel(const float* __restrict__ h, float* __restrict__ mv) {
    const int c = threadIdx.x;
    float s = 0.0f, ss = 0.0f;
    for (int m = 0; m < MPTS; ++m) {
        const float v = h[(size_t)m * DOUT + c];
        s  += v;
        ss += v * v;
    }
    const float mean = s / (float)MPTS;
    const float var  = ss / (float)MPTS - mean * mean;
    mv[c]        = mean;
    mv[DOUT + c] = var;
}

// ---------------------------------------------------------------------------
// Kernel 6: normalize + scale/shift + ReLU -> x_out; also writes n_o.
// ---------------------------------------------------------------------------
__global__ __launch_bounds__(256)
void bn_relu_kernel(const float* __restrict__ h, const float* __restrict__ mv,
                    const float* __restrict__ gamma, const float* __restrict__ beta,
                    float* __restrict__ xout, float* __restrict__ no_out) {
    const int i = blockIdx.x * blockDim.x + threadIdx.x;
    if (i == 0) *no_out = (float)MPTS;
    if (i >= MPTS * DOUT) return;
    const int c = i & (DOUT - 1);
    const float v = (h[i] - mv[c]) * rsqrtf(mv[DOUT + c] + BN_EPS) * gamma[c] + beta[c];
    xout[i] = fmaxf(v, 0.0f);
}

// ---------------------------------------------------------------------------
extern "C" void kernel_launch(void* const* d_in, const int* in_sizes, int n_in,
                              void* d_out, int out_size, void* d_ws, size_t ws_size,
                              hipStream_t stream) {
    const float* p     = (const float*)d_in[0];
    const float* x     = (const float*)d_in[1];
    // d_in[2] = o (unused)
    const float* W     = (const float*)d_in[3];
    // d_in[4] = b (cancels in batch BN)
    const float* gamma = (const float*)d_in[5];
    const float* beta  = (const float*)d_in[6];

    float* out    = (float*)d_out;
    float* np_out = out;                         // [MPTS*3]
    float* x_out  = out + MPTS * 3;              // [MPTS*DOUT]
    float* no_out = out + MPTS * 3 + MPTS * DOUT;// [1]

    char* ws = (char*)d_ws;
    int*   sidx = (int*)ws;                                  ws += (size_t)MPTS * sizeof(int);
    float* feat = (float*)ws;                                ws += (size_t)MPTS * KPAD * sizeof(float);
    float* Wp   = (float*)ws;                                ws += (size_t)KPAD * DOUT * sizeof(float);
    float* h    = (float*)ws;                                ws += (size_t)MPTS * DOUT * sizeof(float);
    float* mv   = (float*)ws;

    fps_kernel<<<1, FPS_T, 0, stream>>>(p, sidx);
    knn_group_kernel<<<MPTS / 256, 256, 0, stream>>>(p, x, sidx, feat, np_out);
    pad_w_kernel<<<(KPAD * DOUT + 255) / 256, 256, 0, stream>>>(W, Wp);
    gemm_wmma_kernel<<<MPTS / 16, 256, 0, stream>>>(feat, Wp, h);
    bn_stats_kernel<<<1, DOUT, 0, stream>>>(h, mv);
    bn_relu_kernel<<<(MPTS * DOUT + 255) / 256, 256, 0, stream>>>(h, mv, gamma, beta, x_out, no_out);
}